// MyModel_61933428411977
// MI455X (gfx1250) — compile-verified
//
#include <hip/hip_runtime.h>
#include <hip/hip_bf16.h>

#define SEQ   4096
#define BATCH 1024
#define IN_F  6
#define HID   3

typedef __attribute__((ext_vector_type(2))) float v2f;
typedef __attribute__((ext_vector_type(8))) float v8f;

__device__ __forceinline__ float fast_tanh(float x) {
#if __has_builtin(__builtin_amdgcn_tanhf)
    return __builtin_amdgcn_tanhf(x);            // v_tanh_f32 (gfx1250 TRANS op)
#elif __has_builtin(__builtin_amdgcn_tanh_f32)
    return __builtin_amdgcn_tanh_f32(x);
#else
    float r;
    asm volatile("v_tanh_f32 %0, %1\n\tv_nop" : "=v"(r) : "v"(x));
    return r;
#endif
}

// Kernel 1: x_proj[t,b,h] = sum_i input[t,b,i] * W_ih[h,i] + b_ih[h] + b_hh[h]
// Flattened [T*B, 6] x [6, 3] GEMM via fp32 WMMA 16x16x4, two K-chunks.
// A layout (ISA 7.12.2, 32-bit A 16x4): lane<16 -> M=lane, K={0,1}; lane>=16 -> M=lane-16, K={2,3}.
// B layout (4x16): VGPR0 = {K=0 | K=2}, VGPR1 = {K=1 | K=3}, N = lane&15
//   (same K-split pattern as the A table and the sparse-B examples in 7.12.4).
// C/D layout: VGPR r -> M = r + 8*(lane>=16), N = lane&15.
__global__ void __launch_bounds__(256)
rnn_xproj_wmma(const float* __restrict__ in, const float* __restrict__ W_ih,
               const float* __restrict__ b_ih, const float* __restrict__ b_hh,
               float* __restrict__ xp)
{
    const int nTiles = (SEQ * BATCH) / 16;
    const int lane   = threadIdx.x & 31;
    const int wave   = (blockIdx.x * blockDim.x + threadIdx.x) >> 5;
    const int nWave  = (gridDim.x * blockDim.x) >> 5;
    const int n      = lane & 15;   // N column (output feature; only 0..2 real)
    const int kh     = lane >> 4;   // K half

    // Lane-resident constants for the whole kernel.
    float b0x = 0.f, b0y = 0.f, b1x = 0.f, b1y = 0.f, bias = 0.f;
    if (n < HID) {
        b0x = W_ih[n * IN_F + 2 * kh];           // B0[k=2kh  ][n] = W_ih[n][2kh]
        b0y = W_ih[n * IN_F + 2 * kh + 1];       // B0[k=2kh+1][n]
        if (kh == 0) {                           // B1 covers i=4,5; rows k=2,3 stay zero
            b1x = W_ih[n * IN_F + 4];
            b1y = W_ih[n * IN_F + 5];
        }
        bias = b_ih[n] + b_hh[n];
    }
    v2f B0; B0.x = b0x; B0.y = b0y;
    v2f B1; B1.x = b1x; B1.y = b1y;

    // Loop-invariant bias accumulator: first WMMA reads it directly as C.
    v8f cbias;
    #pragma unroll
    for (int r = 0; r < 8; ++r) cbias[r] = bias;

    // A1 lane-constant offset: kh==0 lanes fetch i={4,5}; kh==1 lanes re-fetch
    // their own i={2,3} slice (in-bounds) — B1 rows k=2,3 are zero, so the
    // garbage is annihilated. This keeps the load unconditional (no EXEC dance).
    const int o1 = (kh == 0) ? 4 : 2;

    for (int tt = wave; tt < nTiles; tt += nWave) {
        const int rowBase = tt * 16;
        const int row     = rowBase + n;         // A-matrix M for this lane
        const float* rp   = in + row * IN_F;
        v2f A0 = __builtin_nontemporal_load(reinterpret_cast<const v2f*>(rp + 2 * kh));
        v2f A1 = __builtin_nontemporal_load(reinterpret_cast<const v2f*>(rp + o1));

        v8f c;
        c = __builtin_amdgcn_wmma_f32_16x16x4_f32(false, A0, false, B0,
                                                  (short)0, cbias, false, false);
        c = __builtin_amdgcn_wmma_f32_16x16x4_f32(false, A1, false, B1,
                                                  (short)0, c, false, false);

        if (n < HID) {                           // D: M = r + 8*kh, N = n
            float* op = xp + (rowBase + 8 * kh) * HID + n;
            #pragma unroll
            for (int r = 0; r < 8; ++r) op[r * HID] = c[r];
        }
    }
}

// Kernel 2: sequential scan. One lane per batch element; x_proj read/overwritten
// in place inside d_out. Critical path per step: 3-deep FMA + v_tanh_f32.
__global__ void __launch_bounds__(32)
rnn_scan(float* __restrict__ out, const float* __restrict__ W_hh,
         const float* __restrict__ h0in)
{
    const int b = blockIdx.x * blockDim.x + threadIdx.x;
    if (b >= BATCH) return;

    const float w00 = W_hh[0], w01 = W_hh[1], w02 = W_hh[2];
    const float w10 = W_hh[3], w11 = W_hh[4], w12 = W_hh[5];
    const float w20 = W_hh[6], w21 = W_hh[7], w22 = W_hh[8];

    float h0 = h0in[b * 3 + 0], h1 = h0in[b * 3 + 1], h2 = h0in[b * 3 + 2];

    float*    p      = out + b * 3;
    const int stride = BATCH * 3;
    float x0 = p[0], x1 = p[1], x2 = p[2];

    #pragma unroll 4
    for (int t = 0; t < SEQ; ++t) {
        float* np = p + stride;          // at t==SEQ-1 this lands in the h_n tail (valid, unused)
        float y0 = np[0], y1 = np[1], y2 = np[2];   // prefetch next step off the chain

        float a0 = fmaf(w00, h0, fmaf(w01, h1, fmaf(w02, h2, x0)));
        float a1 = fmaf(w10, h0, fmaf(w11, h1, fmaf(w12, h2, x1)));
        float a2 = fmaf(w20, h0, fmaf(w21, h1, fmaf(w22, h2, x2)));
        h0 = fast_tanh(a0);
        h1 = fast_tanh(a1);
        h2 = fast_tanh(a2);

        p[0] = h0; p[1] = h1; p[2] = h2;
        p = np; x0 = y0; x1 = y1; x2 = y2;
    }

    float* hn = out + SEQ * BATCH * 3 + b * 3;
    hn[0] = h0; hn[1] = h1; hn[2] = h2;
}

extern "C" void kernel_launch(void* const* d_in, const int* in_sizes, int n_in,
                              void* d_out, int out_size, void* d_ws, size_t ws_size,
                              hipStream_t stream) {
    (void)in_sizes; (void)n_in; (void)out_size; (void)d_ws; (void)ws_size;
    const float* in   = (const float*)d_in[0];
    const float* h0   = (const float*)d_in[1];
    const float* W_ih = (const float*)d_in[2];
    const float* W_hh = (const float*)d_in[3];
    const float* b_ih = (const float*)d_in[4];
    const float* b_hh = (const float*)d_in[5];
    float* out = (float*)d_out;

    // Stage 1: batched input projection (WMMA f32), writes x_proj into d_out in place.
    rnn_xproj_wmma<<<1024, 256, 0, stream>>>(in, W_ih, b_ih, b_hh, out);
    // Stage 2: serial recurrence, 32 single-wave workgroups x 32 lanes = 1024 chains.
    rnn_scan<<<BATCH / 32, 32, 0, stream>>>(out, W_hh, h0);
}